// QModule_57741540327773
// MI455X (gfx1250) — compile-verified
//
#include <hip/hip_runtime.h>
#include <math.h>

#define C_CH 128
#define L_LEN 4096
#define GN 8
#define N_ROWS (32 * C_CH)  // 4096 (b,c) rows

typedef float v4f __attribute__((ext_vector_type(4)));

// d_ws float layout:
//   [0..4095]      per-row min partials
//   [4096..8191]   per-row max partials
//   [8192..8199]   scale[g]
//   [8200..8207]   zero_point[g]
//   [8208..9231]   A[g*128+c] = softmax(alpha)[g,c] / scale[g]
#define OFF_PMIN 0
#define OFF_PMAX 4096
#define OFF_S    8192
#define OFF_Z    8200
#define OFF_A    8208

// ---------------- kernel 1: per-row (b,c) min/max partials ----------------
// 4096 blocks x 256 threads x 16 floats (4x b128 loads, default RT hint so x
// parks in the 192MB L2 for the second pass)
__global__ void k_reduce(const float* __restrict__ x, float* __restrict__ wsf) {
  const int row = blockIdx.x;
  const float* __restrict__ rp = x + (size_t)row * L_LEN;
  const int tid = threadIdx.x;

  float vmin = __builtin_inff(), vmax = -__builtin_inff();
#pragma unroll
  for (int k = 0; k < 4; ++k) {
    const v4f v = *(const v4f*)(rp + k * 1024 + tid * 4);
    vmin = fminf(vmin, fminf(fminf(v.x, v.y), fminf(v.z, v.w)));
    vmax = fmaxf(vmax, fmaxf(fmaxf(v.x, v.y), fmaxf(v.z, v.w)));
  }
#pragma unroll
  for (int m = 16; m >= 1; m >>= 1) {  // wave32 butterfly (ds permutes)
    vmin = fminf(vmin, __shfl_xor(vmin, m, 32));
    vmax = fmaxf(vmax, __shfl_xor(vmax, m, 32));
  }
  __shared__ float smin[8], smax[8];
  const int wave = tid >> 5, lane = tid & 31;
  if (lane == 0) { smin[wave] = vmin; smax[wave] = vmax; }
  __syncthreads();
  if (tid == 0) {
#pragma unroll
    for (int w = 1; w < 8; ++w) { vmin = fminf(vmin, smin[w]); vmax = fmaxf(vmax, smax[w]); }
    wsf[OFF_PMIN + row] = vmin;
    wsf[OFF_PMAX + row] = vmax;
  }
}

// ---------------- block-wide (128 thr = 4 waves) min/max, result to all ----------------
__device__ __forceinline__ float bred(float v, bool is_max, float* scr, int tid) {
#pragma unroll
  for (int m = 16; m >= 1; m >>= 1) {
    const float o = __shfl_xor(v, m, 32);
    v = is_max ? fmaxf(v, o) : fminf(v, o);
  }
  __syncthreads();  // protect scratch from previous round's readers
  if ((tid & 31) == 0) scr[tid >> 5] = v;
  __syncthreads();
  float r = scr[0];
#pragma unroll
  for (int w = 1; w < 4; ++w) r = is_max ? fmaxf(r, scr[w]) : fminf(r, scr[w]);
  return r;
}

// parallel faithful port of GroupWise_Quantizaion stats (v: one value per thread)
__device__ void gwq_par(float v, bool use_max, float* stats, float* scr, int tid) {
  const float rmin = bred(v, false, scr, tid);
  const float rmax = bred(v, true, scr, tid);
  const float div = rmax - rmin;
  float bounds[GN + 1];
#pragma unroll
  for (int m = 0; m <= GN; ++m) bounds[m] = rmin + (div * (float)m) / (float)GN;
  int mk = 0;  // last containing bin wins (matches overwrite semantics)
#pragma unroll
  for (int j = 0; j < GN; ++j)
    if (v >= bounds[j] && v <= bounds[j + 1]) mk = j + 1;
#pragma unroll
  for (int m = 0; m < GN; ++m) {
    const float pad = use_max ? -__builtin_inff() : __builtin_inff();
    const float cand = (mk == m + 1) ? v : pad;
    const float s = bred(cand, use_max, scr, tid);
    stats[m] = (s == pad) ? bounds[m + 1] : s;  // empty bin -> right edge
  }
}

// ---------------- kernel 2: fold partials + gwq + softmax + coefficients ----------------
// 1 block, 128 threads (one per channel)
__global__ void k_setup(const float* __restrict__ alpha, float* __restrict__ wsf) {
  __shared__ float scr[4];
  const int t = threadIdx.x;

  float mn = __builtin_inff(), mx = -__builtin_inff();
  for (int b = 0; b < 32; ++b) {
    mn = fminf(mn, wsf[OFF_PMIN + b * C_CH + t]);
    mx = fmaxf(mx, wsf[OFF_PMAX + b * C_CH + t]);
  }
  mn = fminf(mn, -4.0f);  // ch_min = where(ch_min > INIT_MIN, INIT_MIN, ch_min)
  mx = fmaxf(mx, 6.0f);   // ch_max = where(ch_max < INIT_MAX, INIT_MAX, ch_max)

  float gmin[GN], gmax[GN];
  gwq_par(mn, /*use_max=*/false, gmin, scr, t);  // mode 'min'
  gwq_par(mx, /*use_max=*/true, gmax, scr, t);   // mode 'max'

  // every thread holds identical gmin/gmax; compute per-group scale/zp locally
  float sc[GN], zp[GN];
#pragma unroll
  for (int g = 0; g < GN; ++g) {
    const float d = fmaxf(gmax[g] - gmin[g], 1e-8f);
    sc[g] = 255.0f / d;                                // n_levels / range
    zp[g] = __builtin_rintf(sc[g] * gmin[g]) + 128.0f; // round(scale*left) + n
  }
  if (t < GN) { wsf[OFF_S + t] = sc[t]; wsf[OFF_Z + t] = zp[t]; }

  // softmax over G for this channel, fused with 1/scale
  float a[GN];
  float am = -__builtin_inff();
#pragma unroll
  for (int g = 0; g < GN; ++g) { a[g] = alpha[g * C_CH + t]; am = fmaxf(am, a[g]); }
  float sum = 0.0f;
#pragma unroll
  for (int g = 0; g < GN; ++g) { a[g] = expf(a[g] - am); sum += a[g]; }
#pragma unroll
  for (int g = 0; g < GN; ++g) wsf[OFF_A + g * C_CH + t] = (a[g] / sum) / sc[g];
}

// ---------------- kernel 3: streaming dequant-mix, per-wave async pipeline ----------------
// Each wave only touches LDS it async-loaded itself (tid-private 16B slots), so
// no workgroup barrier: drain the 4 in-flight DMA chunks with partial
// s_wait_asynccnt and overlap compute with the remaining copies.
__global__ void k_elem(const float* __restrict__ x, float* __restrict__ out,
                       const float* __restrict__ wsf) {
  __shared__ __align__(16) float tile[L_LEN];  // 16 KB
  const int row = blockIdx.x;
  const int c = row & (C_CH - 1);
  const int tid = threadIdx.x;
  const float* __restrict__ rp = x + (size_t)row * L_LEN;
  float* __restrict__ op = out + (size_t)row * L_LEN;

  // issue 4 async b128 copies per lane (each instr = 512B per wave)
#pragma unroll
  for (int k = 0; k < 4; ++k) {
    const int idx = k * 1024 + tid * 4;
    const unsigned lds = (unsigned)(uintptr_t)(&tile[idx]);
    const float* gp = rp + idx;
    asm volatile("global_load_async_to_lds_b128 %0, %1, off"
                 :: "v"(lds), "v"(gp)
                 : "memory");
  }

  // overlap: per-(g,c) coefficients while the DMA is in flight
  float sc[GN], zp[GN], A[GN];
#pragma unroll
  for (int g = 0; g < GN; ++g) {
    sc[g] = wsf[OFF_S + g];
    zp[g] = wsf[OFF_Z + g];
    A[g]  = wsf[OFF_A + g * C_CH + c];
  }

#define PROCESS_CHUNK(K, PENDING)                                              \
  {                                                                            \
    asm volatile("s_wait_asynccnt " #PENDING ::: "memory");                    \
    const int idx = (K) * 1024 + tid * 4;                                      \
    const v4f v = *(const v4f*)(&tile[idx]); /* ds_load_b128 */                \
    v4f o;                                                                     \
    _Pragma("unroll") for (int e = 0; e < 4; ++e) {                            \
      float acc = 0.0f;                                                        \
      _Pragma("unroll") for (int g = 0; g < GN; ++g) {                         \
        float q = __builtin_rintf(__builtin_fmaf(sc[g], v[e], -zp[g]));        \
        q = __builtin_amdgcn_fmed3f(q, -128.0f, 127.0f);                       \
        acc = __builtin_fmaf(A[g], q + zp[g], acc);                            \
      }                                                                        \
      o[e] = acc;                                                              \
    }                                                                          \
    __builtin_nontemporal_store(o, (v4f*)(op + idx)); /* keep x in L2 */       \
  }

  PROCESS_CHUNK(0, 3)
  PROCESS_CHUNK(1, 2)
  PROCESS_CHUNK(2, 1)
  PROCESS_CHUNK(3, 0)
#undef PROCESS_CHUNK
}

extern "C" void kernel_launch(void* const* d_in, const int* in_sizes, int n_in,
                              void* d_out, int out_size, void* d_ws, size_t ws_size,
                              hipStream_t stream) {
  const float* x     = (const float*)d_in[0];  // [32,128,4096] f32
  const float* alpha = (const float*)d_in[1];  // [8,128] f32
  float* out = (float*)d_out;                  // [32,128,4096] f32
  float* wsf = (float*)d_ws;                   // ~37 KB used

  k_reduce<<<N_ROWS, 256, 0, stream>>>(x, wsf);
  k_setup<<<1, C_CH, 0, stream>>>(alpha, wsf);
  k_elem<<<N_ROWS, 256, 0, stream>>>(x, out, wsf);
}